// RelativePositionEncoding_54159537602877
// MI455X (gfx1250) — compile-verified
//
#include <hip/hip_runtime.h>

// ---- problem constants (from reference) ----
#define R_MAX_C   32
#define S_MAX_C   2
#define DIM_OUT   128
#define N_R       (2 * R_MAX_C + 2)                 // 66
#define N_S       (2 * S_MAX_C + 2)                 // 6
#define DIM_IN    (2 * N_R + 1 + N_S)               // 139
#define N_BUCKET  (2 * R_MAX_C + 1)                 // 65 (bucket in [0,64])
#define TBL_N     (N_BUCKET * DIM_OUT)              // 8320 floats = 33,280 B
#define TBL_N4    (TBL_N / 4)                       // 2080 16B chunks

#define THREADS          256                        // 8 wave32s
#define CELLS_PER_BLOCK  1024                       // 8 waves x 128 iters

typedef float        v4f __attribute__((ext_vector_type(4)));
typedef int          v4i __attribute__((ext_vector_type(4)));
typedef int          v8i __attribute__((ext_vector_type(8)));
typedef unsigned int v4u __attribute__((ext_vector_type(4)));
typedef unsigned int u32;

#if defined(__HIP_DEVICE_COMPILE__) && __has_builtin(__builtin_amdgcn_global_load_async_to_lds_b128)
#define HAVE_ASYNC_B128 1
#else
#define HAVE_ASYNC_B128 0
#endif
#if defined(__HIP_DEVICE_COMPILE__) && __has_builtin(__builtin_amdgcn_global_load_async_to_lds_b32)
#define HAVE_ASYNC_B32 1
#else
#define HAVE_ASYNC_B32 0
#endif

// ---------------------------------------------------------------------------
// Kernel 1: precompute table[b][d] = W_token[d,b] + W_pos[d,R_MAX] + w_mask[d]
//           + W_chain[d,2*S_MAX+1] into d_ws (33 KB, contiguous, L2-resident).
// ---------------------------------------------------------------------------
__global__ void __launch_bounds__(THREADS)
rpe_table_kernel(const float* __restrict__ W, float* __restrict__ tbl) {
    const int e = blockIdx.x * THREADS + threadIdx.x;
    if (e >= TBL_N) return;
    const int b = e >> 7;                // bucket
    const int d = e & (DIM_OUT - 1);     // channel
    const float* w = W + (size_t)d * DIM_IN;
    tbl[e] = w[N_R + b] + w[R_MAX_C] + w[2 * N_R] + w[2 * N_R + 1 + (2 * S_MAX_C + 1)];
}

// ---------------------------------------------------------------------------
// Kernel 2: broadcast. One wave32 = one 512B output cell: lane l stores the
// 16B chunk l -> a single coalesced non-temporal global_store_b128 per cell.
// Table staged global->LDS with CDNA5 async b128 loads (ASYNCcnt path).
// ---------------------------------------------------------------------------
__global__ void __launch_bounds__(THREADS)
rpe_broadcast_async_kernel(const float* __restrict__ tbl,
                           const int* __restrict__ seq_ptr,
                           float* __restrict__ out,
                           unsigned ncells) {
    __shared__ v4f table4[TBL_N4];
    const int tid = threadIdx.x;

#if HAVE_ASYNC_B128
    // CDNA5 async global->LDS fill, 16B granules: no VGPR round trip,
    // tracked by ASYNCcnt (global_load_async_to_lds_b128).
    for (int c = tid; c < TBL_N4; c += THREADS) {
        __builtin_amdgcn_global_load_async_to_lds_b128(
            (__attribute__((address_space(1))) v4i*)(tbl + 4 * c),
            (__attribute__((address_space(3))) v4i*)(&table4[c]),
            0, 0);
    }
    __builtin_amdgcn_s_wait_asynccnt(0);
#elif HAVE_ASYNC_B32
    float* table = (float*)table4;
    for (int e = tid; e < TBL_N; e += THREADS) {
        __builtin_amdgcn_global_load_async_to_lds_b32(
            (__attribute__((address_space(1))) int*)(tbl + e),
            (__attribute__((address_space(3))) int*)(&table[e]),
            0, 0);
    }
    __builtin_amdgcn_s_wait_asynccnt(0);
#else
    float* table = (float*)table4;
    for (int e = tid; e < TBL_N; e += THREADS) table[e] = tbl[e];
#endif
    __syncthreads();

    const unsigned seq  = (unsigned)seq_ptr[0];
    const int      wave = tid >> 5;
    const int      lane = tid & 31;

    // Register-cache the two saturated buckets (~94% of all cells): those
    // iterations are a bare NT store with no DS traffic.
    const v4f vlo = table4[lane];                                    // bucket 0
    const v4f vhi = table4[(N_BUCKET - 1) * (DIM_OUT / 4) + lane];   // bucket 64

    unsigned cell = blockIdx.x * CELLS_PER_BLOCK + (unsigned)wave;
    if (cell >= ncells) return;

    // flat cell = (b*seq + i)*seq + j ; batch index irrelevant (broadcast)
    unsigned j = cell % seq;
    unsigned i = (cell / seq) % seq;

    v4f* out4 = (v4f*)out;

    #pragma unroll 4
    for (unsigned it = 0; it < CELLS_PER_BLOCK / 8; ++it) {
        if (cell >= ncells) break;

        const int diff = (int)i - (int)j + R_MAX_C;
        const int b    = diff < 0 ? 0 : (diff > 2 * R_MAX_C ? 2 * R_MAX_C : diff);

        v4f v;
        if (b == 0)                 v = vlo;
        else if (b == 2 * R_MAX_C)  v = vhi;
        else                        v = table4[b * (DIM_OUT / 4) + lane];  // ds_load_b128

        // 512MB write-once stream >> 192MB L2: non-temporal.
        __builtin_nontemporal_store(v, out4 + (size_t)cell * (DIM_OUT / 4) + lane);

        cell += 8;
        j += 8;
        while (j >= seq) { j -= seq; if (++i >= seq) i = 0; }   // no div in hot loop
    }
}

// ---------------------------------------------------------------------------
// Fallback single kernel (no workspace): build table in LDS from W directly.
// ---------------------------------------------------------------------------
__global__ void __launch_bounds__(THREADS)
rpe_broadcast_kernel(const float* __restrict__ W,
                     const int* __restrict__ seq_ptr,
                     float* __restrict__ out,
                     unsigned ncells) {
    __shared__ float cvec[DIM_OUT];
    __shared__ v4f   table4[TBL_N4];
    float* table = (float*)table4;
    const int tid = threadIdx.x;

    if (tid < DIM_OUT) {
        const float* w = W + (size_t)tid * DIM_IN;
        cvec[tid] = w[R_MAX_C] + w[2 * N_R] + w[2 * N_R + 1 + (2 * S_MAX_C + 1)];
    }
    __syncthreads();
    for (int e = tid; e < TBL_N; e += THREADS) {
        const int b = e >> 7, d = e & (DIM_OUT - 1);
        table[e] = W[(size_t)d * DIM_IN + N_R + b] + cvec[d];
    }
    __syncthreads();

    const unsigned seq  = (unsigned)seq_ptr[0];
    const int      wave = tid >> 5, lane = tid & 31;
    const v4f vlo = table4[lane];
    const v4f vhi = table4[(N_BUCKET - 1) * (DIM_OUT / 4) + lane];

    unsigned cell = blockIdx.x * CELLS_PER_BLOCK + (unsigned)wave;
    if (cell >= ncells) return;
    unsigned j = cell % seq;
    unsigned i = (cell / seq) % seq;
    v4f* out4 = (v4f*)out;

    #pragma unroll 4
    for (unsigned it = 0; it < CELLS_PER_BLOCK / 8; ++it) {
        if (cell >= ncells) break;
        const int diff = (int)i - (int)j + R_MAX_C;
        const int b    = diff < 0 ? 0 : (diff > 2 * R_MAX_C ? 2 * R_MAX_C : diff);
        v4f v = (b == 0) ? vlo : (b == 2 * R_MAX_C) ? vhi
                                                    : table4[b * (DIM_OUT / 4) + lane];
        __builtin_nontemporal_store(v, out4 + (size_t)cell * (DIM_OUT / 4) + lane);
        cell += 8; j += 8;
        while (j >= seq) { j -= seq; if (++i >= seq) i = 0; }
    }
}

// ---------------------------------------------------------------------------
// TDM reference (never launched, compiles + lowers on this toolchain):
// 6-arg tensor_load_to_lds with hand-packed D# for a contiguous 8320 x f32
// 1-row tile, completion via s_wait_tensorcnt (TENSORcnt path).
// ---------------------------------------------------------------------------
#if defined(__HIP_DEVICE_COMPILE__) && __has_builtin(__builtin_amdgcn_tensor_load_to_lds)
__global__ void __launch_bounds__(32)
cdna5_tdm_probe(const float* __restrict__ g, float* __restrict__ o) {
    __shared__ v4f stage[TBL_N4];
    const int l = threadIdx.x & 31;

    const u32 lds_addr = (u32)(size_t)(__attribute__((address_space(3))) char*)stage;
    const unsigned long long ga = (unsigned long long)(size_t)g;

    v4u g0;
    g0.x = 1u;                                            // count=1, user mode
    g0.y = lds_addr;                                      // lds_addr [63:32]
    g0.z = (u32)ga;                                       // global_addr low
    g0.w = (u32)((ga >> 32) & 0x01FFFFFFu) | (2u << 30);  // addr[56:32] | type=2

    v8i g1;
    g1[0] = (int)(0x2u << 16);                 // data_size=4B; wg_mask=0; no flags
    g1[1] = (int)((TBL_N & 0xFFFFu) << 16);    // tensor_dim0[15:0] @ bits63:48
    g1[2] = (int)(((TBL_N >> 16) & 0xFFFFu) | (1u << 16)); // tdim0 hi | tensor_dim1=1
    g1[3] = (int)((TBL_N & 0xFFFFu) << 16);    // tdim1 hi=0 | tile_dim0=8320
    g1[4] = 1;                                 // tile_dim1=1 | tile_dim2=0
    g1[5] = (int)TBL_N;                        // tensor_dim0_stride low32
    g1[6] = (int)((TBL_N & 0xFFFFu) << 16);    // stride0 hi=0 | stride1 low16
    g1[7] = 0;                                 // stride1 hi

    v4i gz4 = {0, 0, 0, 0};
    v8i gz8 = {0, 0, 0, 0, 0, 0, 0, 0};
    __builtin_amdgcn_tensor_load_to_lds(g0, g1, gz4, gz4, gz8, 0);
    __builtin_amdgcn_s_wait_tensorcnt(0);
    o[l] = stage[l].x;
}
#endif

// ---------------------------------------------------------------------------
extern "C" void kernel_launch(void* const* d_in, const int* in_sizes, int n_in,
                              void* d_out, int out_size, void* d_ws, size_t ws_size,
                              hipStream_t stream) {
    (void)in_sizes; (void)n_in;
    // inputs: [0]=batch_size(int,1) [1]=seq_len(int,1) [2]=W(float,128*139)
    const int*   d_seq = (const int*)d_in[1];
    const float* d_W   = (const float*)d_in[2];

    const unsigned ncells = (unsigned)(out_size / DIM_OUT);
    const unsigned blocks = (ncells + CELLS_PER_BLOCK - 1) / CELLS_PER_BLOCK;

    if (ws_size >= (size_t)TBL_N * sizeof(float)) {
        rpe_table_kernel<<<(TBL_N + THREADS - 1) / THREADS, THREADS, 0, stream>>>(
            d_W, (float*)d_ws);
        rpe_broadcast_async_kernel<<<blocks, THREADS, 0, stream>>>(
            (const float*)d_ws, d_seq, (float*)d_out, ncells);
    } else {
        rpe_broadcast_kernel<<<blocks, THREADS, 0, stream>>>(
            d_W, d_seq, (float*)d_out, ncells);
    }
}